// PSRoIAlign_72069551227028
// MI455X (gfx1250) — compile-verified
//
#include <hip/hip_runtime.h>
#include <cstdint>

// ---------------- problem constants (match reference setup_inputs) ----------
namespace psroi {
constexpr int N    = 8;
constexpr int C    = 490;
constexpr int H    = 100;
constexpr int W    = 100;
constexpr int PH   = 7;
constexpr int PW   = 7;
constexpr int BINS = PH * PW;          // 49
constexpr int COUT = C / BINS;         // 10
constexpr float SCALE = 0.0625f;
constexpr int WHALF = 50;              // W split per transpose block
}

typedef unsigned int u32x4 __attribute__((ext_vector_type(4)));
typedef int          i32x8 __attribute__((ext_vector_type(8)));
typedef int          i32x4 __attribute__((ext_vector_type(4)));

// ---------------------------------------------------------------------------
// Phase 1: layout transform  in[N][C][H][W]  ->  ws[N][bin][H][W][COUT]
// One block per (n, h, w-half). TDM DMAs the 490x50 tile (row stride H*W)
// into LDS with a single tensor_load_to_lds; waves then emit the permuted
// tile with fully coalesced stores (cout fastest -> contiguous runs).
// ---------------------------------------------------------------------------
__global__ void psroi_transpose(const float* __restrict__ in,
                                float* __restrict__ dst) {
  using namespace psroi;
  extern __shared__ float smem[];                  // C * WHALF floats (98 KB)

  const int blk  = blockIdx.x;
  const int half = blk & 1;
  const int nh   = blk >> 1;
  const int n    = nh / H;
  const int h    = nh % H;
  const int w0   = half * WHALF;

  // element offset of (n, c=0, h, w0)
  const unsigned long long base_el =
      ((unsigned long long)n * C * H + (unsigned long long)h) * W + w0;

#if defined(__gfx1250__) && __has_builtin(__builtin_amdgcn_tensor_load_to_lds)
  if (threadIdx.x < 32) {                          // wave 0 only (wave32)
    const unsigned long long ga =
        (unsigned long long)(uintptr_t)in + base_el * 4ull;

    u32x4 g0;
    g0[0] = 1u;                                    // count=1, user descriptor
    g0[1] = 0u;                                    // lds_addr = 0 (dyn LDS base)
    g0[2] = (unsigned)(ga & 0xFFFFFFFFu);          // global_addr[31:0]
    g0[3] = (unsigned)((ga >> 32) & 0x01FFFFFFu)   // global_addr[56:32]
            | (2u << 30);                          // type = 2 ("image")

    i32x8 g1;
    g1[0] = 0x00020000;                            // data_size = 2 (4 bytes)
    g1[1] = (int)((unsigned)WHALF << 16);          // tensor_dim0[15:0]
    g1[2] = (int)((unsigned)C << 16);              // dim0 hi=0 | tensor_dim1 lo
    g1[3] = (int)((unsigned)WHALF << 16);          // dim1 hi=0 | tile_dim0=50
    g1[4] = (int)C;                                // tile_dim1=490, tile_dim2=0
    g1[5] = (int)(H * W);                          // tensor_dim0_stride lo32
    g1[6] = 0;                                     // stride0 hi | stride1 lo
    g1[7] = 0;

    i32x4 z4 = {0, 0, 0, 0};
    i32x8 z8 = {0, 0, 0, 0, 0, 0, 0, 0};
    // 6-arg form (clang-23 / therock-10.0 headers): groups 2/3 disabled (zero).
    __builtin_amdgcn_tensor_load_to_lds(g0, g1, z4, z4, z8, 0);
    __builtin_amdgcn_s_wait_tensorcnt(0);
  }
  __syncthreads();
#else
  for (int f = threadIdx.x; f < C * WHALF; f += blockDim.x) {
    const int c = f / WHALF, wc = f % WHALF;
    smem[f] = in[base_el + (unsigned long long)c * (H * W) + wc];
  }
  __syncthreads();
#endif

  // Emit permuted tile: for each (bin, w, cout) -> dst[n][bin][h][w][cout].
  // Flat order = cout fastest, then w -> contiguous 500-float runs per bin.
  for (int f = threadIdx.x; f < BINS * WHALF * COUT; f += blockDim.x) {
    const int bin  = f / (WHALF * COUT);
    const int rem  = f % (WHALF * COUT);
    const int wi   = rem / COUT;
    const int cout = rem % COUT;
    const float v  = smem[(cout * BINS + bin) * WHALF + wi];
    const unsigned long long off =
        (((unsigned long long)(n * BINS + bin)) * H + h) * (W * COUT)
        + (unsigned long long)(w0 + wi) * COUT + cout;
    dst[off] = v;
  }
}

// ---------------------------------------------------------------------------
// Bilinear index/weight logic (mirrors torchvision / reference).
// ---------------------------------------------------------------------------
__device__ __forceinline__ bool bparams(float coord, int size,
                                        int& lo, int& hi, float& frac) {
  const bool valid = (coord >= -1.0f) && (coord <= (float)size);
  float c = fmaxf(coord, 0.0f);
  int low = (int)floorf(c);
  if (low > size - 1) low = size - 1;
  int high = low + 1;
  if (high > size - 1) high = size - 1;
  if (low >= size - 1) c = (float)low;
  frac = c - (float)low;
  lo = low; hi = high;
  return valid;
}

// ---------------------------------------------------------------------------
// Phase 2: gather from transformed layout. One thread per (k, bin, cout-pair);
// each bilinear corner is a single float2 load of 2 adjacent couts.
// ---------------------------------------------------------------------------
__global__ void psroi_gather(const float* __restrict__ rois,
                             const float* __restrict__ feat,
                             float* __restrict__ out, int K) {
  using namespace psroi;
  const int tid = blockIdx.x * blockDim.x + threadIdx.x;
  const int total = K * BINS * (COUT / 2);
  if (tid >= total) return;

  const int pair  = tid % (COUT / 2);
  const int binId = tid / (COUT / 2);
  const int k   = binId / BINS;
  const int bin = binId % BINS;
  const int ph  = bin / PW;
  const int pw  = bin % PW;

  const float* r = rois + k * 5;
  const int   b  = (int)r[0];
  const float sw = r[1] * SCALE - 0.5f;
  const float sh = r[2] * SCALE - 0.5f;
  const float ew = r[3] * SCALE - 0.5f;
  const float eh = r[4] * SCALE - 0.5f;
  const float bin_h = fmaxf(eh - sh, 0.1f) * (1.0f / PH);
  const float bin_w = fmaxf(ew - sw, 0.1f) * (1.0f / PW);

  const float2* f2 = (const float2*)feat;
  const int rb = (b * BINS + bin) * H;   // row base in (b,bin) plane

  float accx = 0.0f, accy = 0.0f;
#pragma unroll
  for (int s = 0; s < 4; ++s) {
    const int sy = s >> 1, sx = s & 1;
    const float yy = sh + ((float)ph + ((float)sy + 0.5f) * 0.5f) * bin_h;
    const float xx = sw + ((float)pw + ((float)sx + 0.5f) * 0.5f) * bin_w;

    int yl, yh2, xl, xh2; float ly, lx;
    const bool vy = bparams(yy, H, yl, yh2, ly);
    const bool vx = bparams(xx, W, xl, xh2, lx);
    if (!(vy && vx)) continue;
    const float hy = 1.0f - ly, hx = 1.0f - lx;

    const int i00 = ((rb + yl)  * W + xl)  * (COUT / 2) + pair;
    const int i01 = ((rb + yl)  * W + xh2) * (COUT / 2) + pair;
    const int i10 = ((rb + yh2) * W + xl)  * (COUT / 2) + pair;
    const int i11 = ((rb + yh2) * W + xh2) * (COUT / 2) + pair;
    const float2 v00 = f2[i00];
    const float2 v01 = f2[i01];
    const float2 v10 = f2[i10];
    const float2 v11 = f2[i11];

    const float w00 = hy * hx, w01 = hy * lx, w10 = ly * hx, w11 = ly * lx;
    accx += w00 * v00.x + w01 * v01.x + w10 * v10.x + w11 * v11.x;
    accy += w00 * v00.y + w01 * v01.y + w10 * v10.y + w11 * v11.y;
  }

  const int cbase = k * COUT + pair * 2;             // out[K][COUT][PH][PW]
  out[cbase * BINS + bin]       = accx * 0.25f;
  out[(cbase + 1) * BINS + bin] = accy * 0.25f;
}

// ---------------------------------------------------------------------------
// Fallback: direct gather from native NCHW layout (used if ws too small).
// One thread per output element; 16 scalar gathers each.
// ---------------------------------------------------------------------------
__global__ void psroi_direct(const float* __restrict__ in,
                             const float* __restrict__ rois,
                             float* __restrict__ out, int K) {
  using namespace psroi;
  const int tid = blockIdx.x * blockDim.x + threadIdx.x;
  const int total = K * COUT * BINS;
  if (tid >= total) return;

  const int k    = tid / (COUT * BINS);
  const int rrem = tid % (COUT * BINS);
  const int cout = rrem / BINS;
  const int bin  = rrem % BINS;
  const int ph   = bin / PW;
  const int pw   = bin % PW;

  const float* r = rois + k * 5;
  const int   b  = (int)r[0];
  const float sw = r[1] * SCALE - 0.5f;
  const float sh = r[2] * SCALE - 0.5f;
  const float ew = r[3] * SCALE - 0.5f;
  const float eh = r[4] * SCALE - 0.5f;
  const float bin_h = fmaxf(eh - sh, 0.1f) * (1.0f / PH);
  const float bin_w = fmaxf(ew - sw, 0.1f) * (1.0f / PW);

  const int c = cout * BINS + bin;
  const int base = (b * C + c) * H * W;

  float acc = 0.0f;
#pragma unroll
  for (int s = 0; s < 4; ++s) {
    const int sy = s >> 1, sx = s & 1;
    const float yy = sh + ((float)ph + ((float)sy + 0.5f) * 0.5f) * bin_h;
    const float xx = sw + ((float)pw + ((float)sx + 0.5f) * 0.5f) * bin_w;

    int yl, yh2, xl, xh2; float ly, lx;
    const bool vy = bparams(yy, H, yl, yh2, ly);
    const bool vx = bparams(xx, W, xl, xh2, lx);
    if (!(vy && vx)) continue;
    const float hy = 1.0f - ly, hx = 1.0f - lx;

    acc += hy * hx * in[base + yl  * W + xl ]
         + hy * lx * in[base + yl  * W + xh2]
         + ly * hx * in[base + yh2 * W + xl ]
         + ly * lx * in[base + yh2 * W + xh2];
  }
  out[tid] = acc * 0.25f;
}

// ---------------------------------------------------------------------------
extern "C" void kernel_launch(void* const* d_in, const int* in_sizes, int n_in,
                              void* d_out, int out_size, void* d_ws, size_t ws_size,
                              hipStream_t stream) {
  using namespace psroi;
  const float* in   = (const float*)d_in[0];
  const float* rois = (const float*)d_in[1];
  float* out        = (float*)d_out;
  const int K = in_sizes[1] / 5;

  const size_t need = (size_t)N * BINS * H * W * COUT * sizeof(float); // 156.8 MB

  if (ws_size >= need) {
    float* ws = (float*)d_ws;
    // Phase 1: TDM-staged layout transform. 2 blocks per (n,h), 98 KB dyn LDS.
    const int g1 = N * H * 2;
    psroi_transpose<<<g1, 256, C * WHALF * (int)sizeof(float), stream>>>(in, ws);
    // Phase 2: vectorized gather.
    const int total = K * BINS * (COUT / 2);
    psroi_gather<<<(total + 255) / 256, 256, 0, stream>>>(rois, ws, out, K);
  } else {
    const int total = K * COUT * BINS;
    psroi_direct<<<(total + 255) / 256, 256, 0, stream>>>(in, rois, out, K);
  }
}